// UniformScatter_31980326486571
// MI455X (gfx1250) — compile-verified
//
#include <hip/hip_runtime.h>
#include <hip/hip_bf16.h>

// Pure streaming copy (uniform gate == 1.0, contiguous reshape) on the CDNA5
// async global<->LDS DMA path. Key trick: the async ops add INST_OFFSET to
// BOTH the global and the LDS address (CDNA5 ISA), so one
// VGPR address pair + one LDS-offset VGPR serves a whole 32 KiB batch via
// offset:k*4096 immediates. Non-temporal hints keep the 512 MiB stream from
// churning the 192 MB L2.

#define TPB   256   // threads per block (8 waves of 32)
#define BATCH 8     // tiles per batch; tile = TPB lanes * 16B = 4 KiB

// One async load: copies 16B/lane from global address PTR + OFFSTR to LDS address LDSREG + OFFSTR.
#define AL(LDSREG, PTR, OFFSTR)                                                \
    asm volatile("global_load_async_to_lds_b128 %0, %1, off offset:" OFFSTR    \
                 " th:TH_LOAD_NT"                                              \
                 :: "v"(LDSREG), "v"(PTR) : "memory")

// One async store: copies 16B/lane from LDS address LDSREG + OFFSTR to global address PTR + OFFSTR.
#define AS(PTR, LDSREG, OFFSTR)                                                \
    asm volatile("global_store_async_from_lds_b128 %0, %1, off offset:" OFFSTR \
                 " th:TH_STORE_NT"                                             \
                 :: "v"(PTR), "v"(LDSREG) : "memory")

#define AL8(LDSREG, PTR) do {                                                  \
    AL(LDSREG, PTR, "0");     AL(LDSREG, PTR, "4096");                         \
    AL(LDSREG, PTR, "8192");  AL(LDSREG, PTR, "12288");                        \
    AL(LDSREG, PTR, "16384"); AL(LDSREG, PTR, "20480");                        \
    AL(LDSREG, PTR, "24576"); AL(LDSREG, PTR, "28672"); } while (0)

#define AS8(PTR, LDSREG) do {                                                  \
    AS(PTR, LDSREG, "0");     AS(PTR, LDSREG, "4096");                         \
    AS(PTR, LDSREG, "8192");  AS(PTR, LDSREG, "12288");                        \
    AS(PTR, LDSREG, "16384"); AS(PTR, LDSREG, "20480");                        \
    AS(PTR, LDSREG, "24576"); AS(PTR, LDSREG, "28672"); } while (0)

__global__ __launch_bounds__(TPB)
void UniformScatter_async_copy_kernel(const float4* __restrict__ in,
                                      float4* __restrict__ out,
                                      unsigned nbatch) {
    // Two LDS buffers of BATCH tiles each: 2 * 8 * 256 * 16B = 64 KiB.
    __shared__ float4 buf[2][BATCH * TPB];

    const unsigned lane = threadIdx.x;
    // Low 32 bits of the flat shared-aperture address == LDS byte offset.
    const unsigned ldsA = (unsigned)(size_t)(&buf[0][0]) + lane * 16u;
    const unsigned ldsB = (unsigned)(size_t)(&buf[1][0]) + lane * 16u;

    unsigned b = blockIdx.x;
    if (b >= nbatch) return;

    const size_t batchVec = (size_t)BATCH * TPB;   // float4s per batch

    // ---- prologue: async-load batch b into buffer A -------------------------
    AL8(ldsA, in + (size_t)b * batchVec + lane);

    unsigned curLds = ldsA, nxtLds = ldsB;
    unsigned cur = b;

    // ---- steady state: store(cur) overlaps load(next) -----------------------
    for (unsigned nxt = b + gridDim.x; nxt < nbatch; nxt += gridDim.x) {
        // Loads of batch `cur` (and stores of the previous batch) must be done
        // before we read/overwrite the LDS buffers. Partial waits are unsafe:
        // loads vs stores retire out of order on the shared ASYNCcnt.
        asm volatile("s_wait_asynccnt 0x0" ::: "memory");

        AS8(out + (size_t)cur * batchVec + lane, curLds);
        AL8(nxtLds, in + (size_t)nxt * batchVec + lane);

        cur = nxt;
        unsigned t = curLds; curLds = nxtLds; nxtLds = t;
    }

    // ---- epilogue: drain last batch -----------------------------------------
    asm volatile("s_wait_asynccnt 0x0" ::: "memory");
    AS8(out + (size_t)cur * batchVec + lane, curLds);
    asm volatile("s_wait_asynccnt 0x0" ::: "memory");
}

// Scalar tail for any remainder not covered by full 32 KiB batches
// (zero for the 16384x4096 reference shape, kept for generality).
__global__ __launch_bounds__(TPB)
void UniformScatter_tail_kernel(const float* __restrict__ in,
                                float* __restrict__ out,
                                size_t n) {
    size_t i = (size_t)blockIdx.x * TPB + threadIdx.x;
    if (i < n) out[i] = in[i];
}

extern "C" void kernel_launch(void* const* d_in, const int* in_sizes, int n_in,
                              void* d_out, int out_size, void* d_ws, size_t ws_size,
                              hipStream_t stream) {
    const float* in  = (const float*)d_in[0];
    float*       out = (float*)d_out;

    const size_t total    = (size_t)out_size;       // floats (== in_sizes[0])
    const size_t nvec     = total / 4;              // float4s
    const size_t batchVec = (size_t)BATCH * TPB;    // float4s per batch
    const size_t nbatch   = nvec / batchVec;        // full 32 KiB batches

    if (nbatch > 0) {
        // ~4 batches per block; plenty of blocks to cover all WGPs.
        unsigned blocks = (unsigned)((nbatch + 3) / 4);
        if (blocks > 8192u) blocks = 8192u;
        UniformScatter_async_copy_kernel<<<blocks, TPB, 0, stream>>>(
            (const float4*)in, (float4*)out, (unsigned)nbatch);
    }

    const size_t doneFloats = nbatch * batchVec * 4;
    if (doneFloats < total) {
        const size_t rem = total - doneFloats;
        unsigned blocks = (unsigned)((rem + TPB - 1) / TPB);
        UniformScatter_tail_kernel<<<blocks, TPB, 0, stream>>>(
            in + doneFloats, out + doneFloats, rem);
    }
}